// YOLO_Loss_80547816669457
// MI455X (gfx1250) — compile-verified
//
#include <hip/hip_runtime.h>
#include <hip/hip_bf16.h>
#include <math.h>

// ---------------- problem constants ----------------
#define NUM_CLASSES 80
#define BATCH 16
#define TCAP  50
#define NA    3
#define LH    76
#define LW    76
#define HW    (LH*LW)              // 5776
#define NCELL (BATCH*NA*HW)        // 277248
#define PRED_CH 255                // NA * 85
#define IGNORE_THR 0.5f

// scaled anchors: ANCHORS / stride, stride = 608/76 = 8
__device__ __constant__ float c_aw[3] = {14.5f, 19.5f, 46.625f};
__device__ __constant__ float c_ah[3] = {11.25f, 24.75f, 40.75f};

// ---------------- workspace layout (bytes) ----------------
#define FLAGS_OFF    0u            // NCELL bytes (u8): noobj_mask zeroed-cells
#define NPOS_OFF     277248u       // 1 int
#define REC_OFF      278528u       // 800 records * 12 floats = 38400 B
#define POSPART_OFF  317440u       // 6 * NBP floats
#define CONFPART_OFF 318464u       // NBC floats
#define NBP 7                      // blocks in positive kernel (128 thr each; 896 >= 800)
#define NBC 256                    // blocks in conf kernel (256 thr each)

__device__ __forceinline__ float sigmoidf(float x) { return 1.0f / (1.0f + __expf(-x)); }
__device__ __forceinline__ float clip_log(float p) { return fmaxf(__logf(p), -100.0f); }
__device__ __forceinline__ float bce(float p, float t) {
    return -(t * clip_log(p) + (1.0f - t) * clip_log(1.0f - p));
}

// ---------------- kernel 1: encode targets (tiny, serial per batch => deterministic) ---
__global__ void k_encode(const float* __restrict__ annot, float* __restrict__ ws_f,
                         unsigned char* __restrict__ flags, int* __restrict__ npos) {
    int b = threadIdx.x;
    if (b >= BATCH) return;
    unsigned char vld[TCAP];
    int   cell[TCAP], rA[TCAP], rGi[TCAP], rGj[TCAP], rC[TCAP];
    float rTx[TCAP], rTy[TCAP], rTw[TCAP], rTh[TCAP];

    for (int t = 0; t < TCAP; ++t) {
        const float* a = annot + (b * TCAP + t) * 5;
        float a0 = a[0], a1 = a[1], a2 = a[2], a3 = a[3], a4 = a[4];
        bool valid = (a0 + a1 + a2 + a3 + a4) != 0.0f;
        float gx = a1 * (float)LW, gy = a2 * (float)LH;
        float gw = a3 * (float)LW, gh = a4 * (float)LH;
        int gi = (int)floorf(gx), gj = (int)floorf(gy);
        float best = -1.0f; int bn = 0; float ious[3];
        for (int k = 0; k < 3; ++k) {
            float inter = fminf(gw, c_aw[k]) * fminf(gh, c_ah[k]);
            float uni = gw * (gh + 1e-9f) + c_aw[k] * (c_ah[k] + 1e-9f) - inter + 1e-9f;
            float iou = inter / uni; ious[k] = iou;
            if (iou > best) { best = iou; bn = k; }   // first max wins (jnp.argmax)
        }
        bool inb = (gi >= 0) && (gi < LW) && (gj >= 0) && (gj < LH);
        if (valid && inb) {
            for (int k = 0; k < 3; ++k)
                if (ious[k] > IGNORE_THR)
                    flags[(b * NA + k) * HW + gj * LW + gi] = 1;   // noobj_mask -> 0
        }
        vld[t] = (valid && inb) ? 1 : 0;
        cell[t] = (bn * LH + gj) * LW + gi;
        rA[t] = bn; rGi[t] = gi; rGj[t] = gj;
        rTx[t] = gx - (float)gi;  rTy[t] = gy - (float)gj;
        rTw[t] = __logf(gw / c_aw[bn] + 1e-16f);
        rTh[t] = __logf(gh / c_ah[bn] + 1e-16f);
        rC[t] = (int)a0;
    }
    // last-writer-wins dedup (matches sequential scatter .set semantics); n_pos = #distinct cells
    int cnt = 0;
    for (int t = 0; t < TCAP; ++t) {
        bool win = (vld[t] != 0);
        if (win) {
            for (int u = t + 1; u < TCAP; ++u)
                if (vld[u] && cell[u] == cell[t]) { win = false; break; }
        }
        if (win) cnt++;
        float* r = ws_f + (REC_OFF / 4) + (b * TCAP + t) * 12;
        r[0] = win ? 1.0f : 0.0f;
        r[1] = (float)rA[t]; r[2] = (float)rGj[t]; r[3] = (float)rGi[t];
        r[4] = rTx[t]; r[5] = rTy[t]; r[6] = rTw[t]; r[7] = rTh[t];
        r[8] = (float)rC[t];
    }
    atomicAdd(npos, cnt);   // integer atomic: deterministic value
}

// ---------------- kernel 2: gather positive cells, 6 loss partial sums ----------------
__global__ void k_pos(const float* __restrict__ pred, const float* __restrict__ ws_f,
                      float* __restrict__ posPart) {
    int r = blockIdx.x * blockDim.x + threadIdx.x;
    float sx = 0, sy = 0, sw = 0, sh = 0, sc = 0, scl = 0;
    if (r < BATCH * TCAP) {
        const float* rec = ws_f + (REC_OFF / 4) + r * 12;
        if (rec[0] != 0.0f) {
            int b = r / TCAP;
            int a = (int)rec[1], gj = (int)rec[2], gi = (int)rec[3];
            float tx = rec[4], ty = rec[5], tw = rec[6], th = rec[7];
            int cls = (int)rec[8];
            const float* base = pred + (size_t)(b * PRED_CH + a * 85) * HW + gj * LW + gi;
            float p0 = base[0], p1 = base[HW], p2 = base[2 * HW], p3 = base[3 * HW], p4 = base[4 * HW];
            sx = bce(sigmoidf(p0), tx);
            sy = bce(sigmoidf(p1), ty);
            sw = (p2 - tw) * (p2 - tw);
            sh = (p3 - th) * (p3 - th);
            sc = -clip_log(sigmoidf(p4));                 // BCE(conf, 1)
            for (int c = 0; c < NUM_CLASSES; ++c) {
                float pc = sigmoidf(base[(size_t)(5 + c) * HW]);
                scl += bce(pc, (c == cls) ? 1.0f : 0.0f);
            }
        }
    }
    __shared__ float sdata[128];
    float vals[6] = { sx, sy, sw, sh, sc, scl };
    for (int k = 0; k < 6; ++k) {
        sdata[threadIdx.x] = vals[k];
        __syncthreads();
        for (int s = 64; s > 0; s >>= 1) {
            if (threadIdx.x < s) sdata[threadIdx.x] += sdata[threadIdx.x + s];
            __syncthreads();
        }
        if (threadIdx.x == 0) posPart[k * NBP + blockIdx.x] = sdata[0];
        __syncthreads();
    }
}

// ---------------- kernel 3: dense no-object conf term (only the 48 conf planes) -------
__global__ void k_conf(const float* __restrict__ pred, const unsigned char* __restrict__ flags,
                       float* __restrict__ confPart) {
    int tid = blockIdx.x * blockDim.x + threadIdx.x;
    int stride = gridDim.x * blockDim.x;
    float s = 0.0f;
    for (int i = tid; i < NCELL; i += stride) {
        int b  = i / (NA * HW);
        int rm = i - b * (NA * HW);
        int a  = rm / HW;
        int hw = rm - a * HW;
        float logit = pred[(size_t)(b * PRED_CH + a * 85 + 4) * HW + hw];
        if (!flags[i]) {
            float conf = sigmoidf(logit);
            s += -clip_log(1.0f - conf);                  // BCE(conf*noobj, 0)
        }
    }
    __shared__ float sdata[256];
    sdata[threadIdx.x] = s;
    __syncthreads();
    for (int st = 128; st > 0; st >>= 1) {
        if (threadIdx.x < st) sdata[threadIdx.x] += sdata[threadIdx.x + st];
        __syncthreads();
    }
    if (threadIdx.x == 0) confPart[blockIdx.x] = sdata[0];
}

// ---------------- kernel 4: single-wave finisher; WMMA f32 row-sum reduction ----------
typedef float v8f __attribute__((ext_vector_type(8)));
typedef float v2f __attribute__((ext_vector_type(2)));

__global__ void k_final(const float* __restrict__ ws_f, const int* __restrict__ npos,
                        float* __restrict__ out) {
    const float* confPart = ws_f + (CONFPART_OFF / 4);
    const float* posPart  = ws_f + (POSPART_OFF / 4);
    int lane = threadIdx.x;
    float total_conf = 0.0f;
#if __has_builtin(__builtin_amdgcn_wmma_f32_16x16x4_f32)
    // Reduce 256 partials: A(16x4) = 64 partials per chunk, B(4x16) = ones -> row sums,
    // accumulate 4 chunks into C. Exact f32 FMA math; EXEC all-ones (32 threads, no branch).
    v8f acc = {};
    v2f ones; ones.x = 1.0f; ones.y = 1.0f;
    for (int chunk = 0; chunk < 4; ++chunk) {
        v2f a;
        a.x = confPart[chunk * 64 + lane * 2 + 0];
        a.y = confPart[chunk * 64 + lane * 2 + 1];
        acc = __builtin_amdgcn_wmma_f32_16x16x4_f32(
            /*neg_a=*/false, a, /*neg_b=*/false, ones,
            /*c_mod=*/(short)0, acc, /*reuse_a=*/false, /*reuse_b=*/false);
    }
    // every row-sum is replicated across all 16 columns -> total = sum(all D)/16
    float s = acc[0] + acc[1] + acc[2] + acc[3] + acc[4] + acc[5] + acc[6] + acc[7];
    for (int off = 16; off > 0; off >>= 1) s += __shfl_xor(s, off, 32);
    total_conf = s * (1.0f / 16.0f);
#else
    if (lane == 0) {
        float s = 0.0f;
        for (int i = 0; i < NBC; ++i) s += confPart[i];
        total_conf = s;
    }
#endif
    if (lane == 0) {
        float S[6];
        for (int k = 0; k < 6; ++k) {
            float s = 0.0f;
            for (int i = 0; i < NBP; ++i) s += posPart[k * NBP + i];
            S[k] = s;
        }
        float Nf = (float)NCELL;
        float np = (float)(*npos);
        float loss = 0.5f * (S[0] + S[1]) / Nf          // LAMBDA_XY * (loss_x + loss_y)
                   + 2.5f * (S[2] + S[3]) / Nf          // LAMBDA_WH * (loss_w + loss_h)
                   + (S[4] + 0.5f * total_conf) / Nf    // loss_conf
                   + S[5] / (np * (float)NUM_CLASSES);  // loss_cls
        out[0] = loss;
    }
}

// ---------------- launch ----------------
extern "C" void kernel_launch(void* const* d_in, const int* in_sizes, int n_in,
                              void* d_out, int out_size, void* d_ws, size_t ws_size,
                              hipStream_t stream) {
    (void)in_sizes; (void)n_in; (void)out_size; (void)ws_size;
    const float* pred  = (const float*)d_in[0];
    const float* annot = (const float*)d_in[1];
    float* ws_f = (float*)d_ws;
    unsigned char* flags = (unsigned char*)d_ws + FLAGS_OFF;
    int* npos = (int*)((char*)d_ws + NPOS_OFF);
    float* posPart  = (float*)((char*)d_ws + POSPART_OFF);
    float* confPart = (float*)((char*)d_ws + CONFPART_OFF);

    // zero the noobj flag grid + n_pos counter every call (graph-capture safe)
    hipMemsetAsync(d_ws, 0, NPOS_OFF + 4, stream);

    k_encode<<<1, 32, 0, stream>>>(annot, ws_f, flags, npos);
    k_pos  <<<NBP, 128, 0, stream>>>(pred, ws_f, posPart);
    k_conf <<<NBC, 256, 0, stream>>>(pred, flags, confPart);
    k_final<<<1, 32, 0, stream>>>(ws_f, npos, (float*)d_out);
}